// SAGE_32804960207226
// MI455X (gfx1250) — compile-verified
//
#include <hip/hip_runtime.h>

typedef __attribute__((ext_vector_type(2))) float v2f;
typedef __attribute__((ext_vector_type(4))) float v4f;
typedef __attribute__((ext_vector_type(8))) float v8f;

// ---------------- zero scratch ----------------
__global__ void sage_zero_kernel(float* __restrict__ p, long n) {
    long i = (long)blockIdx.x * blockDim.x + threadIdx.x;
    long stride = (long)gridDim.x * blockDim.x;
    for (; i < n; i += stride) p[i] = 0.0f;
}

// ---------------- weight repack into WMMA-native layout ----------------
// Wp[((nt*(F_in/2) + kk)*16 + ln)*2 + j] = W[(2*kk+j)*F_out_src + nt*16+ln]  (0 if col padded)
__global__ void sage_packw_kernel(const float* __restrict__ W, float* __restrict__ Wp,
                                  int F_in, int F_out_src, int F_out_dst) {
    long total = (long)F_in * F_out_dst;
    long i = (long)blockIdx.x * blockDim.x + threadIdx.x;
    long stride = (long)gridDim.x * blockDim.x;
    int half = F_in >> 1;
    for (; i < total; i += stride) {
        int j  = (int)(i & 1);
        int ln = (int)((i >> 1) & 15);
        long r = i >> 5;
        int kk = (int)(r % half);
        int nt = (int)(r / half);
        int k  = 2 * kk + j;
        int col = nt * 16 + ln;
        Wp[i] = (col < F_out_src) ? W[(long)k * F_out_src + col] : 0.0f;
    }
}

__global__ void sage_packb_kernel(const float* __restrict__ b, float* __restrict__ bp,
                                  int C_src, int C_dst) {
    int i = blockIdx.x * blockDim.x + threadIdx.x;
    if (i < C_dst) bp[i] = (i < C_src) ? b[i] : 0.0f;
}

// ---------------- scatter mean accumulate ----------------
__global__ void sage_scatter_kernel(const float* __restrict__ hsrc,
                                    const int* __restrict__ src,
                                    const int* __restrict__ dst,
                                    float* __restrict__ agg,
                                    float* __restrict__ cnt,
                                    int E, int F) {
    const int qpf = F >> 2;
    long t = (long)blockIdx.x * blockDim.x + threadIdx.x;
    long total = (long)E * qpf;
    if (t >= total) return;
    int e = (int)(t / qpf);
    int q = (int)(t % qpf);
    int s = src[e], d = dst[e];
    const v4f* hp = (const v4f*)(hsrc + (long)s * F);
    v4f v = hp[q];
    float* ap = agg + (long)d * F + (long)q * 4;
    atomicAdd(ap + 0, v.x);
    atomicAdd(ap + 1, v.y);
    atomicAdd(ap + 2, v.z);
    atomicAdd(ap + 3, v.w);
    if (q == 0) atomicAdd(cnt + d, 1.0f);
}

// ---------------- fused SAGE layer ----------------
// out = (agg/max(cnt,1))@Wl + bl + hdst@Wr [+relu], F_out % 16 == 0 (pre-padded).
// Each wave computes a 16 x (16*NT) block: A fragment loaded once per k-step,
// NT independent V_WMMA_F32_16X16X4_F32 accumulators. Weights pre-packed.
template <int NT>
__global__ __launch_bounds__(128) void sage_wmma_kernel(
        const float* __restrict__ agg,
        const float* __restrict__ cnt,
        const float* __restrict__ hdst,
        const float* __restrict__ Wlp,   // packed
        const float* __restrict__ bl,    // padded to F_out
        const float* __restrict__ Wrp,   // packed
        float* __restrict__ out,
        int n_dst, int F_in, int F_out, int relu) {
    const int lane = threadIdx.x & 31;
    const int wave = threadIdx.x >> 5;
    const int ntiles_m = n_dst >> 4;
    const int groups_n = F_out / (16 * NT);
    int gw = blockIdx.x * (blockDim.x >> 5) + wave;      // wave-uniform
    if (gw >= ntiles_m * groups_n) return;                // wave-uniform exit
    const int tm = gw / groups_n;
    const int g  = gw % groups_n;
    const int n0 = g * (16 * NT);
    const int nt0 = n0 >> 4;

    const int ln   = lane & 15;
    const int hi   = lane >> 4;            // 0 for lanes 0-15, 1 for 16-31
    const int mrow = (tm << 4) + ln;       // A: lane holds row M = lane&15
    const int koff = hi << 1;              // lanes 16-31 hold K+2,K+3

    const float cv  = cnt[mrow];
    const float inv = 1.0f / fmaxf(cv, 1.0f);

    const long tile_stride = (long)F_in * 16;   // floats per packed N-tile

    v8f acc[NT];
    #pragma unroll
    for (int t = 0; t < NT; ++t) acc[t] = (v8f){};

    // pass 1: (agg * inv) @ Wl
    {
        const float* arow = agg + (long)mrow * F_in;
        for (int k = 0; k < F_in; k += 4) {
            v2f a = *(const v2f*)(arow + k + koff);
            a.x *= inv; a.y *= inv;
            const long boff = ((long)((k >> 1) + hi) * 16 + ln) * 2;
            #pragma unroll
            for (int t = 0; t < NT; ++t) {
                v2f b = *(const v2f*)(Wlp + (long)(nt0 + t) * tile_stride + boff);
                acc[t] = __builtin_amdgcn_wmma_f32_16x16x4_f32(
                    false, a, false, b, (short)0, acc[t], false, false);
            }
        }
    }
    // pass 2: hdst @ Wr
    {
        const float* hrow = hdst + (long)mrow * F_in;
        for (int k = 0; k < F_in; k += 4) {
            v2f a = *(const v2f*)(hrow + k + koff);
            const long boff = ((long)((k >> 1) + hi) * 16 + ln) * 2;
            #pragma unroll
            for (int t = 0; t < NT; ++t) {
                v2f b = *(const v2f*)(Wrp + (long)(nt0 + t) * tile_stride + boff);
                acc[t] = __builtin_amdgcn_wmma_f32_16x16x4_f32(
                    false, a, false, b, (short)0, acc[t], false, false);
            }
        }
    }

    // epilogue: bias + relu + store (C/D: VGPR v -> row v, +8 for hi lanes, col = lane&15)
    const int rbase = (tm << 4) + (hi << 3);
    #pragma unroll
    for (int t = 0; t < NT; ++t) {
        const int col = n0 + t * 16 + ln;
        const float bias = bl[col];
        #pragma unroll
        for (int v = 0; v < 8; ++v) {
            float val = acc[t][v] + bias;
            if (relu) val = fmaxf(val, 0.0f);
            out[(long)(rbase + v) * F_out + col] = val;
        }
    }
}

// ---------------- log_softmax (reads padded h, writes h + lsm) ----------------
__global__ void sage_logsoftmax_kernel(const float* __restrict__ hp,
                                       float* __restrict__ out_h,
                                       float* __restrict__ out_lsm,
                                       int R, int C, int stride) {
    int row = blockIdx.x * (blockDim.x >> 5) + (threadIdx.x >> 5);
    int lane = threadIdx.x & 31;
    if (row >= R) return;
    const float* hr = hp + (long)row * stride;
    float m = -3.402823466e38f;
    for (int c = lane; c < C; c += 32) m = fmaxf(m, hr[c]);
    for (int off = 16; off; off >>= 1) m = fmaxf(m, __shfl_xor(m, off, 32));
    float s = 0.0f;
    for (int c = lane; c < C; c += 32) s += expf(hr[c] - m);
    for (int off = 16; off; off >>= 1) s += __shfl_xor(s, off, 32);
    float lg = logf(s);
    for (int c = lane; c < C; c += 32) {
        float v = hr[c];
        out_h[(long)row * C + c] = v;
        out_lsm[(long)row * C + c] = v - m - lg;
    }
}

extern "C" void kernel_launch(void* const* d_in, const int* in_sizes, int n_in,
                              void* d_out, int out_size, void* d_ws, size_t ws_size,
                              hipStream_t stream) {
    const float* x    = (const float*)d_in[0];
    const int* src0   = (const int*)d_in[1];
    const int* dst0   = (const int*)d_in[2];
    const int* src1   = (const int*)d_in[3];
    const int* dst1   = (const int*)d_in[4];
    const int* src2   = (const int*)d_in[5];
    const int* dst2   = (const int*)d_in[6];
    const float* Wl0  = (const float*)d_in[7];
    const float* bl0  = (const float*)d_in[8];
    const float* Wr0  = (const float*)d_in[9];
    const float* Wl1  = (const float*)d_in[10];
    const float* bl1  = (const float*)d_in[11];
    const float* Wr1  = (const float*)d_in[12];
    const float* Wl2  = (const float*)d_in[13];
    const float* bl2  = (const float*)d_in[14];
    const float* Wr2  = (const float*)d_in[15];
    float* out = (float*)d_out;

    const long N1 = 102400, N2 = 10240, N3 = 1024;
    const int  F0 = 128, F1 = 256, F2 = 256, F3 = 47, F3p = 48;
    const int  E0 = 1024000, E1 = 102400, E2 = 10240;

    // workspace carving (floats; all sizes even -> v2f alignment holds)
    float* ws = (float*)d_ws;
    long o = 0;
    float* agg0 = ws + o; o += N1 * F0;
    float* cnt0 = ws + o; o += N1;
    float* agg1 = ws + o; o += N2 * F1;
    float* cnt1 = ws + o; o += N2;
    float* agg2 = ws + o; o += N3 * F2;
    float* cnt2 = ws + o; o += N3;
    const long zeroN = o;                  // contiguous agg+cnt region
    float* h1   = ws + o; o += N1 * F1;
    float* h2   = ws + o; o += N2 * F2;
    float* h3p  = ws + o; o += N3 * F3p;   // padded layer-2 output
    float* Wlp0 = ws + o; o += (long)F0 * F1;
    float* Wrp0 = ws + o; o += (long)F0 * F1;
    float* Wlp1 = ws + o; o += (long)F1 * F2;
    float* Wrp1 = ws + o; o += (long)F1 * F2;
    float* Wlp2 = ws + o; o += (long)F2 * F3p;
    float* Wrp2 = ws + o; o += (long)F2 * F3p;
    float* blp2 = ws + o; o += F3p;

    // zero agg/cnt scratch every call (graph-replay safe)
    sage_zero_kernel<<<4096, 256, 0, stream>>>(ws, zeroN);

    // repack weights (cheap; deterministic per call)
    sage_packw_kernel<<<256, 256, 0, stream>>>(Wl0, Wlp0, F0, F1, F1);
    sage_packw_kernel<<<256, 256, 0, stream>>>(Wr0, Wrp0, F0, F1, F1);
    sage_packw_kernel<<<256, 256, 0, stream>>>(Wl1, Wlp1, F1, F2, F2);
    sage_packw_kernel<<<256, 256, 0, stream>>>(Wr1, Wrp1, F1, F2, F2);
    sage_packw_kernel<<<64, 256, 0, stream>>>(Wl2, Wlp2, F2, F3, F3p);
    sage_packw_kernel<<<64, 256, 0, stream>>>(Wr2, Wrp2, F2, F3, F3p);
    sage_packb_kernel<<<1, 64, 0, stream>>>(bl2, blp2, F3, F3p);

    // ---- layer 0 ----
    {
        long total = (long)E0 * (F0 >> 2);
        sage_scatter_kernel<<<(int)((total + 255) / 256), 256, 0, stream>>>(
            x, src0, dst0, agg0, cnt0, E0, F0);
        int waves = (int)(N1 >> 4) * (F1 / (16 * 8));
        sage_wmma_kernel<8><<<(waves + 3) / 4, 128, 0, stream>>>(
            agg0, cnt0, x, Wlp0, bl0, Wrp0, h1, (int)N1, F0, F1, 1);
    }
    // ---- layer 1 ----
    {
        long total = (long)E1 * (F1 >> 2);
        sage_scatter_kernel<<<(int)((total + 255) / 256), 256, 0, stream>>>(
            h1, src1, dst1, agg1, cnt1, E1, F1);
        int waves = (int)(N2 >> 4) * (F2 / (16 * 8));
        sage_wmma_kernel<8><<<(waves + 3) / 4, 128, 0, stream>>>(
            agg1, cnt1, h1, Wlp1, bl1, Wrp1, h2, (int)N2, F1, F2, 1);
    }
    // ---- layer 2 (padded output to h3p) ----
    {
        long total = (long)E2 * (F2 >> 2);
        sage_scatter_kernel<<<(int)((total + 255) / 256), 256, 0, stream>>>(
            h2, src2, dst2, agg2, cnt2, E2, F2);
        int waves = (int)(N3 >> 4) * (F3p / (16 * 3));
        sage_wmma_kernel<3><<<(waves + 3) / 4, 128, 0, stream>>>(
            agg2, cnt2, h2, Wlp2, blp2, Wrp2, h3p, (int)N3, F2, F3p, 0);
    }
    // ---- h + log_softmax into d_out ----
    {
        int waves_per_block = 256 / 32;
        int blocks = (int)((N3 + waves_per_block - 1) / waves_per_block);
        sage_logsoftmax_kernel<<<blocks, 256, 0, stream>>>(
            h3p, out, out + N3 * F3, (int)N3, F3, F3p);
    }
}